// HashEmbeddingMRUGrid_69148973466029
// MI455X (gfx1250) — compile-verified
//
#include <hip/hip_runtime.h>
#include <cmath>

// InstantNGP hash-grid embedding, forward. 1M points x 16 levels x F=2.
// Gather/latency bound: tables (64MB) fit in MI455X 192MB L2; levels 0-1
// (~100KB incl. edge overrun) are staged into the 320KB/WGP LDS with
// GLOBAL_LOAD_ASYNC_TO_LDS (ASYNCcnt). Output/coords use non-temporal
// hints so the 140MB of streaming traffic doesn't evict the tables from L2.

#define NLVL 16
#define TBL (1u << 19)          // 2^19 entries per level
// Reference indexes up to res + res*res + res*res*res for linear levels
// (corner offsets are not clamped), so stage that many entries.
#define LDS_N0 4369             // 16 + 16^2 + 16^3 + 1
#define LDS_N1 8421             // 20 + 20^2 + 20^3 + 1
#define LDS_TOT (LDS_N0 + LDS_N1)

typedef float v4f __attribute__((ext_vector_type(4)));

struct Cfg {
    int res[NLVL];
    unsigned hashMask;          // bit l set -> level l uses xor spatial hash
    int B;
};

__device__ __forceinline__ void async_copy_b64_to_lds(const void* gsrc, void* ldst) {
    // GV-mode async copy: each lane copies 8 bytes from the global address in vaddr to the LDS address in vdst, tracked by ASYNCcnt.
    unsigned loff = (unsigned)(size_t)ldst;                 // low 32b = LDS offset
    unsigned long long gaddr = (unsigned long long)(size_t)gsrc;
    asm volatile("global_load_async_to_lds_b64 %0, %1, off"
                 :: "v"(loff), "v"(gaddr) : "memory");
}

__global__ __launch_bounds__(256)
void hashgrid_kernel(const float* __restrict__ x,
                     const float* __restrict__ tables,
                     float* __restrict__ out,
                     Cfg cfg) {
    __shared__ float2 lds01[LDS_TOT];   // ~100KB of the 320KB/WGP LDS

    // Stage level-0 and level-1 tables into LDS via the async copy engine.
    const float2* tab0 = (const float2*)tables;
    const float2* tab1 = (const float2*)(tables + (size_t)TBL * 2u);
    for (int i = threadIdx.x; i < LDS_N0; i += 256)
        async_copy_b64_to_lds(tab0 + i, &lds01[i]);
    for (int i = threadIdx.x; i < LDS_N1; i += 256)
        async_copy_b64_to_lds(tab1 + i, &lds01[LDS_N0 + i]);
    asm volatile("s_wait_asynccnt 0x0" ::: "memory");       // my slice landed
    __syncthreads();                                        // everyone's slice landed

    const int stride = gridDim.x * blockDim.x;
    for (int t = blockIdx.x * blockDim.x + threadIdx.x; t < cfg.B; t += stride) {
        // Prefetch the next point's coordinates (global_prefetch_b8).
        if (t + stride < cfg.B)
            __builtin_prefetch(&x[3 * (t + stride)], 0, 1);

        // Streaming reads: non-temporal so they don't pollute L2 table residency.
        const float px = __builtin_nontemporal_load(&x[3 * t + 0]);
        const float py = __builtin_nontemporal_load(&x[3 * t + 1]);
        const float pz = __builtin_nontemporal_load(&x[3 * t + 2]);

        float acc[2 * NLVL];

        #pragma unroll
        for (int l = 0; l < NLVL; ++l) {
            const int res = cfg.res[l];
            const float rs = 0.5f * (float)res;     // 1/cell
            const float fx = (px + 1.0f) * rs;
            const float fy = (py + 1.0f) * rs;
            const float fz = (pz + 1.0f) * rs;
            int ix = __float2int_rd(fx);            // pos >= 0: floor
            int iy = __float2int_rd(fy);
            int iz = __float2int_rd(fz);
            ix = ix < 0 ? 0 : (ix > res - 1 ? res - 1 : ix);
            iy = iy < 0 ? 0 : (iy > res - 1 ? res - 1 : iy);
            iz = iz < 0 ? 0 : (iz > res - 1 ? res - 1 : iz);
            const float wx = fx - (float)ix;
            const float wy = fy - (float)iy;
            const float wz = fz - (float)iz;

            // Issue all 8 gathers up front for memory-level parallelism.
            float2 v[8];
            if (l < 2) {
                const int base = (l == 0) ? 0 : LDS_N0;
                #pragma unroll
                for (int c = 0; c < 8; ++c) {
                    const int cx = ix + ((c >> 2) & 1);
                    const int cy = iy + ((c >> 1) & 1);
                    const int cz = iz + (c & 1);
                    const int idx = cx + res * (cy + res * cz);
                    v[c] = lds01[base + idx];             // ds_load_b64
                }
            } else {
                const float2* tab = (const float2*)tables + (size_t)l * TBL;
                const bool useHash = (cfg.hashMask >> l) & 1u;
                #pragma unroll
                for (int c = 0; c < 8; ++c) {
                    const int cx = ix + ((c >> 2) & 1);
                    const int cy = iy + ((c >> 1) & 1);
                    const int cz = iz + (c & 1);
                    unsigned idx;
                    if (useHash) {
                        idx = (unsigned)cx
                            ^ ((unsigned)cy * 2654435761u)
                            ^ ((unsigned)cz * 805459861u);
                        idx &= (TBL - 1u);
                    } else {
                        idx = (unsigned)(cx + res * (cy + res * cz));
                    }
                    v[c] = tab[idx];                      // global_load_b64 gather (L2 hit)
                }
            }

            const float wx0 = 1.0f - wx;
            const float wy0 = 1.0f - wy;
            const float wz0 = 1.0f - wz;
            const float wyz[4] = { wy0 * wz0, wy0 * wz, wy * wz0, wy * wz };

            float o0 = 0.0f, o1 = 0.0f;
            #pragma unroll
            for (int c = 0; c < 8; ++c) {
                const float w = (((c >> 2) & 1) ? wx : wx0) * wyz[c & 3];
                o0 = fmaf(w, v[c].x, o0);
                o1 = fmaf(w, v[c].y, o1);
            }
            acc[2 * l]     = o0;
            acc[2 * l + 1] = o1;
        }

        // 128B per lane, 8 x b128 non-temporal stores (output never re-read).
        v4f* o4 = (v4f*)(out + (size_t)t * (2 * NLVL));
        const v4f* a4 = (const v4f*)acc;
        #pragma unroll
        for (int k = 0; k < (2 * NLVL) / 4; ++k)
            __builtin_nontemporal_store(a4[k], &o4[k]);
    }
}

extern "C" void kernel_launch(void* const* d_in, const int* in_sizes, int n_in,
                              void* d_out, int out_size, void* d_ws, size_t ws_size,
                              hipStream_t stream) {
    const float* x      = (const float*)d_in[0];
    const float* tables = (const float*)d_in[1];
    float* out          = (float*)d_out;

    Cfg cfg;
    const double factor = exp((log(512.0) - log(16.0)) / 15.0);
    unsigned mask = 0;
    for (int i = 0; i < NLVL; ++i) {
        const int r = (int)floor(16.0 * pow(factor, (double)i));
        cfg.res[i] = r;
        const double r3 = (double)r * (double)r * (double)r;
        if (r3 > (double)TBL) mask |= (1u << i);
    }
    cfg.hashMask = mask;
    cfg.B = in_sizes[0] / 3;

    const int blocks = 1024;   // ~4 points/thread; amortizes LDS staging
    hipLaunchKernelGGL(hashgrid_kernel, dim3(blocks), dim3(256), 0, stream,
                       x, tables, out, cfg);
}